// GCMCConv_38010460570190
// MI455X (gfx1250) — compile-verified
//
#include <hip/hip_runtime.h>

typedef __attribute__((ext_vector_type(2))) float v2f;
typedef __attribute__((ext_vector_type(8))) float v8f;

#define IN_DIM 128
#define OUT_DIM 32

// ---------------------------------------------------------------------------
// Kernel 1: h = X @ W^T + b, then wh = h * cj, via V_WMMA_F32_16X16X4_F32.
// One wave -> one 16-row tile, two 16x16 WMMA accumulators cover OUT_DIM=32.
// ---------------------------------------------------------------------------
__global__ void __launch_bounds__(256) gcmc_gemm_cj_kernel(
    const float* __restrict__ X,     // [N, 128]
    const float* __restrict__ W,     // [32, 128]
    const float* __restrict__ bias,  // [32]
    const float* __restrict__ cj,    // [N]
    float* __restrict__ wh,          // [N, 32] (workspace)
    int n_nodes)
{
    const int wave   = (blockIdx.x * blockDim.x + threadIdx.x) >> 5;  // wave-uniform
    const int ntiles = (n_nodes + 15) >> 4;
    if (wave >= ntiles) return;        // whole-wave exit: EXEC stays all-ones for WMMA

    const int lane = threadIdx.x & 31;
    const int m    = lane & 15;        // A row / B column within tile
    const int kh   = lane >> 4;        // K sub-pair selector {0,1}<->{2,3}

    int arow = wave * 16 + m;
    if (arow >= n_nodes) arow = n_nodes - 1;   // clamp (N%16==0 in practice)

    const float* xrow = X + (size_t)arow * IN_DIM + 2 * kh;        // A frag base
    const float* w0   = W + (size_t)m        * IN_DIM + 2 * kh;    // B frag, n = m
    const float* w1   = W + (size_t)(m + 16) * IN_DIM + 2 * kh;    // B frag, n = m+16

    v8f acc0 = {};
    v8f acc1 = {};

#pragma unroll
    for (int kb = 0; kb < IN_DIM; kb += 4) {
        v2f a  = *(const v2f*)(xrow + kb);   // A[m][kb + 2*kh + {0,1}]
        v2f b0 = *(const v2f*)(w0 + kb);     // W[n0][kb + 2*kh + {0,1}]
        v2f b1 = *(const v2f*)(w1 + kb);
        // (neg_a, A, neg_b, B, c_mod, C, reuse_a, reuse_b)
        acc0 = __builtin_amdgcn_wmma_f32_16x16x4_f32(
            false, a, false, b0, (short)0, acc0, false, false);
        acc1 = __builtin_amdgcn_wmma_f32_16x16x4_f32(
            false, a, false, b1, (short)0, acc1, false, false);
    }

    // C/D layout: lane holds column n = lane&15; VGPR r holds row m = r + 8*kh.
    const int   n0 = m, n1 = m + 16;
    const float bias0 = bias[n0];
    const float bias1 = bias[n1];

#pragma unroll
    for (int r = 0; r < 8; ++r) {
        const int mo   = r + 8 * kh;
        const int orow = wave * 16 + mo;
        if (orow < n_nodes) {
            const float s = cj[orow];
            wh[(size_t)orow * OUT_DIM + n0] = (acc0[r] + bias0) * s;
            wh[(size_t)orow * OUT_DIM + n1] = (acc1[r] + bias1) * s;
        }
    }
}

// ---------------------------------------------------------------------------
// Kernel 2: zero the output accumulator (d_out is poisoned by the harness).
// ---------------------------------------------------------------------------
__global__ void __launch_bounds__(256) gcmc_zero_kernel(float4* __restrict__ out, int n4)
{
    const int i = blockIdx.x * blockDim.x + threadIdx.x;
    if (i < n4) out[i] = make_float4(0.f, 0.f, 0.f, 0.f);
}

// ---------------------------------------------------------------------------
// Kernel 3: edge scatter-sum. 8 threads per edge, 4 floats per thread.
// float4 gather from wh[src], 4x global_atomic_add_f32 into out[dst].
// ---------------------------------------------------------------------------
__global__ void __launch_bounds__(256) gcmc_scatter_kernel(
    const float* __restrict__ wh,
    const int*   __restrict__ edge_src,
    const int*   __restrict__ edge_dst,
    float*       __restrict__ out,
    int n_edges)
{
    const long long t = (long long)blockIdx.x * blockDim.x + threadIdx.x;
    const long long e = t >> 3;
    if (e >= n_edges) return;
    const int c = ((int)t & 7) * 4;

    const int s = edge_src[e];
    const int d = edge_dst[e];

    const float4 v = *(const float4*)(wh + (size_t)s * OUT_DIM + c);
    float* o = out + (size_t)d * OUT_DIM + c;
    unsafeAtomicAdd(o + 0, v.x);   // hardware global_atomic_add_f32
    unsafeAtomicAdd(o + 1, v.y);
    unsafeAtomicAdd(o + 2, v.z);
    unsafeAtomicAdd(o + 3, v.w);
}

// ---------------------------------------------------------------------------
// Kernel 4: out *= ci (destination degree scaling). 8 threads per node.
// ---------------------------------------------------------------------------
__global__ void __launch_bounds__(256) gcmc_ci_kernel(
    float* __restrict__ out, const float* __restrict__ ci, int n_nodes)
{
    const int t    = blockIdx.x * blockDim.x + threadIdx.x;
    const int node = t >> 3;
    if (node >= n_nodes) return;
    const int c = (t & 7) * 4;

    const float s = ci[node];
    float4* p = (float4*)(out + (size_t)node * OUT_DIM + c);
    float4  v = *p;
    v.x *= s; v.y *= s; v.z *= s; v.w *= s;
    *p = v;
}

// ---------------------------------------------------------------------------
extern "C" void kernel_launch(void* const* d_in, const int* in_sizes, int n_in,
                              void* d_out, int out_size, void* d_ws, size_t ws_size,
                              hipStream_t stream)
{
    const float* X    = (const float*)d_in[0];  // src_feats [N,128]
    const float* cj   = (const float*)d_in[1];  // [N]
    const float* ci   = (const float*)d_in[2];  // [N]
    const float* W    = (const float*)d_in[3];  // [32,128]
    const float* bias = (const float*)d_in[4];  // [32]
    const int*   esrc = (const int*)d_in[5];    // [E]
    const int*   edst = (const int*)d_in[6];    // [E]

    float* out = (float*)d_out;                 // [N,32]
    float* wh  = (float*)d_ws;                  // [N,32] scratch (12.8 MB)

    const int n_nodes = in_sizes[1];            // cj has N elements
    const int n_edges = in_sizes[5];

    // 1) WMMA GEMM + cj scaling -> wh
    {
        const int ntiles = (n_nodes + 15) / 16;       // one wave per 16-row tile
        const int blocks = (ntiles + 7) / 8;          // 8 waves per 256-thread block
        gcmc_gemm_cj_kernel<<<blocks, 256, 0, stream>>>(X, W, bias, cj, wh, n_nodes);
    }

    // 2) zero output accumulator
    {
        const int n4 = out_size / 4;
        gcmc_zero_kernel<<<(n4 + 255) / 256, 256, 0, stream>>>((float4*)out, n4);
    }

    // 3) scatter-sum over edges
    {
        const long long threads = (long long)n_edges * 8;
        const int blocks = (int)((threads + 255) / 256);
        gcmc_scatter_kernel<<<blocks, 256, 0, stream>>>(wh, esrc, edst, out, n_edges);
    }

    // 4) destination degree scaling
    {
        const int threads = n_nodes * 8;
        gcmc_ci_kernel<<<(threads + 255) / 256, 256, 0, stream>>>(out, ci, n_nodes);
    }
}